// NGramRepeatBlock_55336358642958
// MI455X (gfx1250) — compile-verified
//
#include <hip/hip_runtime.h>
#include <hip/hip_bf16.h>
#include <stdint.h>

// ---------------------------------------------------------------------------
// NGramRepeatBlock (n = 3), MI455X / gfx1250.
//
// Roofline: out is B*vocab f32 (~103 MB). Work = 103 MB read + 103 MB write
// + ~4 MB token reads => ~210 MB @ 23.3 TB/s ~= 9 us. Memory bound; no
// matrix math exists in this op, so WMMA is not applicable. We instead use
// the CDNA5 async global->LDS data mover path to stage each token row into
// LDS once (the n-gram scan reads every token ~3x), and b128 vector loads
// for the streaming copy.
// ---------------------------------------------------------------------------

#define NGRAM 3
#define MAX_LDS_TOKENS 4096   // 16 KB of int32 token staging per block

#if defined(__gfx1250__) && __has_builtin(__builtin_amdgcn_global_load_async_to_lds_b128)
#define USE_ASYNC_LDS 1
// Builtin signature (from hipcc diagnostic): param0 = int4 addrspace(1)*,
// param1 = int4 addrspace(3)*, then two integer immediates (offset, cpol).
typedef int v4i __attribute__((ext_vector_type(4)));
typedef __attribute__((address_space(1))) v4i* gptr_v4i;
typedef __attribute__((address_space(3))) v4i* lptr_v4i;
#else
#define USE_ASYNC_LDS 0
#endif

__device__ __forceinline__ void wait_asynccnt0() {
#if USE_ASYNC_LDS
#if __has_builtin(__builtin_amdgcn_s_wait_asynccnt)
  __builtin_amdgcn_s_wait_asynccnt(0);
#else
  asm volatile("s_wait_asynccnt 0" ::: "memory");
#endif
#endif
}

// ---------------------------------------------------------------------------
// Kernel 1: streaming copy lprobs -> out, float4 (global_load_b128 /
// global_store_b128). One element-quad per thread; tiny tail handled by the
// first (total & 3) threads.
// ---------------------------------------------------------------------------
__global__ __launch_bounds__(256) void ngram_copy_kernel(
    const float* __restrict__ src, float* __restrict__ dst, int total) {
  const int n4 = total >> 2;
  const int i  = blockIdx.x * blockDim.x + threadIdx.x;
  if (i < n4) {
    reinterpret_cast<float4*>(dst)[i] =
        reinterpret_cast<const float4*>(src)[i];
  }
  const int tail = total & 3;
  if (i < tail) {
    const int base = n4 << 2;
    dst[base + i] = src[base + i];
  }
}

// ---------------------------------------------------------------------------
// Kernel 2: per-row n-gram scan. bsz/beam/step are device scalars, so all
// shape math happens on-GPU; the grid-stride over rows tolerates any B.
// Token row is staged into LDS via the async-to-LDS path when possible.
// ---------------------------------------------------------------------------
__global__ __launch_bounds__(256) void ngram_scan_kernel(
    const int* __restrict__ tokens, float* __restrict__ out,
    const int* __restrict__ p_bsz, const int* __restrict__ p_beam,
    const int* __restrict__ p_step, int lprobs_total) {
  __shared__ int sTok[MAX_LDS_TOKENS];

  const int bsz    = p_bsz[0];
  const int beam   = p_beam[0];
  const int step   = p_step[0];
  const int B      = bsz * beam;
  if (B <= 0) return;
  const int seqLen = step + 1;
  const int vocab  = lprobs_total / B;
  const int csp    = step + 2 - NGRAM;  // check_start_pos
  if (csp <= 0) return;                 // reference: return lprobs unchanged

  const int tid  = threadIdx.x;
  const int nthr = blockDim.x;

  for (int row = blockIdx.x; row < B; row += gridDim.x) {
    const int* __restrict__ tok    = tokens + (size_t)row * (size_t)seqLen;
    float* __restrict__     outRow = out    + (size_t)row * (size_t)vocab;

    if (seqLen <= MAX_LDS_TOKENS) {
      __syncthreads();  // guard sTok reuse across row iterations
#if USE_ASYNC_LDS
      if ((seqLen & 3) == 0) {
        // Async global->LDS, 16 B per transfer, per-lane addresses.
        const int n4 = seqLen >> 2;
        for (int c = tid; c < n4; c += nthr) {
          __builtin_amdgcn_global_load_async_to_lds_b128(
              (gptr_v4i)(tok + 4 * c), (lptr_v4i)(sTok + 4 * c), 0, 0);
        }
        wait_asynccnt0();  // each wave drains its own ASYNCcnt
      } else
#endif
      {
        for (int i = tid; i < seqLen; i += nthr) sTok[i] = tok[i];
      }
      __syncthreads();   // all waves' staged data now visible

      const int last0 = sTok[step - 1];
      const int last1 = sTok[step];
      for (int s = tid; s < csp; s += nthr) {
        if (sTok[s] == last0 && sTok[s + 1] == last1) {
          const int banned = sTok[s + 2];
          if ((unsigned)banned < (unsigned)vocab)
            outRow[banned] = -__builtin_inff();  // benign write race: same value
        }
      }
    } else {
      // Fallback for rows too large for LDS staging: scan from global.
      const int last0 = tok[step - 1];
      const int last1 = tok[step];
      for (int s = tid; s < csp; s += nthr) {
        const int a = tok[s];
        const int b = tok[s + 1];
        if (a == last0 && b == last1) {
          const int banned = tok[s + 2];
          if ((unsigned)banned < (unsigned)vocab)
            outRow[banned] = -__builtin_inff();
        }
      }
    }
  }
}

// ---------------------------------------------------------------------------
// Launch. Inputs (dict order): tokens(int), lprobs(f32), bsz, beam_size, step
// (device scalars, int). Output: f32, same shape as lprobs.
// Copy first, scan second; stream order serializes them.
// ---------------------------------------------------------------------------
extern "C" void kernel_launch(void* const* d_in, const int* in_sizes, int n_in,
                              void* d_out, int out_size, void* d_ws, size_t ws_size,
                              hipStream_t stream) {
  const int*   tokens = (const int*)d_in[0];
  const float* lprobs = (const float*)d_in[1];
  const int*   p_bsz  = (const int*)d_in[2];
  const int*   p_beam = (const int*)d_in[3];
  const int*   p_step = (const int*)d_in[4];
  float*       out    = (float*)d_out;

  const int total   = out_size;
  const int threads = 256;
  int blocks = ((total >> 2) + threads - 1) / threads;
  if (blocks < 1) blocks = 1;
  ngram_copy_kernel<<<blocks, threads, 0, stream>>>(lprobs, out, total);

  // 512 blocks (one WGP-resident block per beam row at B=512); grid-stride
  // inside the kernel handles any B read from the device scalars.
  ngram_scan_kernel<<<512, threads, 0, stream>>>(tokens, out, p_bsz, p_beam,
                                                 p_step, total);
}